// decoder_5128190951936
// MI455X (gfx1250) — compile-verified
//
#include <hip/hip_runtime.h>
#include <hip/hip_bf16.h>

typedef __attribute__((ext_vector_type(2))) float v2f;
typedef __attribute__((ext_vector_type(8))) float v8f;

#define FEAT 128          // output feature width of both GCN layers
#define BN_EPS 1e-5f

// ---------------------------------------------------------------------------
// Degree / normalization
// ---------------------------------------------------------------------------
__global__ void deg_kernel(const int* __restrict__ col, float* __restrict__ deg, int n_edges) {
    int e = blockIdx.x * blockDim.x + threadIdx.x;
    if (e < n_edges) atomicAdd(&deg[col[e]], 1.0f);
}

__global__ void dis_kernel(const float* __restrict__ deg, float* __restrict__ dis, int n) {
    int i = blockIdx.x * blockDim.x + threadIdx.x;
    if (i < n) dis[i] = rsqrtf(deg[i] + 1.0f);
}

// ---------------------------------------------------------------------------
// WMMA GEMM: C[n,128] = A[n,K] * B[K,128], fp32 via V_WMMA_F32_16X16X4_F32.
// Block = 256 threads (8 waves). Block owns one 16-row M tile (A staged in
// LDS via b128 loads; +4 row padding -> conflict-free & 16B aligned); wave w
// owns output columns [16w, 16w+16). Per ISA 7.12.2 layouts:
//   A 16x4 : lanes 0-15 -> {K=kb,kb+1}, lanes 16-31 -> {K=kb+2,kb+3}, M=lane&15
//   B 4x16 : mirrored over N
//   C/D    : VGPR r -> M = r + 8*(lane>>4), N = lane&15
// ---------------------------------------------------------------------------
template <int K>
__global__ __launch_bounds__(256)
void gemm_wmma_kernel(const float* __restrict__ A, const float* __restrict__ B,
                      float* __restrict__ C, int nrows) {
    __shared__ float As[16][K + 4];

    const int mbase = blockIdx.x * 16;
    const bool fullTile = (mbase + 16 <= nrows);

    // cooperative A-tile staging, float4 granularity (16*K/4 vectors)
    if (fullTile) {
        for (int i = threadIdx.x; i < 4 * K; i += 256) {       // 4*K float4s
            int e = i * 4;
            int r = e / K, c = e - r * K;
            float4 v = *reinterpret_cast<const float4*>(A + (size_t)(mbase + r) * K + c);
            *reinterpret_cast<float4*>(&As[r][c]) = v;
        }
    } else {
        for (int i = threadIdx.x; i < 16 * K; i += 256) {
            int r = i / K, c = i - r * K;
            int grow = mbase + r;
            As[r][c] = (grow < nrows) ? A[(size_t)grow * K + c] : 0.0f;
        }
    }
    __syncthreads();

    const int wave = threadIdx.x >> 5;
    const int lane = threadIdx.x & 31;
    const int half = lane >> 4;     // which K pair this lane supplies
    const int l15  = lane & 15;
    const int ncol = wave * 16 + l15;

    v8f acc = {};
#pragma unroll
    for (int kb = 0; kb < K; kb += 4) {
        const int ka = kb + half * 2;
        v2f av; av.x = As[l15][ka];
                av.y = As[l15][ka + 1];
        v2f bv; bv.x = B[(size_t)ka * FEAT + ncol];
                bv.y = B[(size_t)(ka + 1) * FEAT + ncol];
        acc = __builtin_amdgcn_wmma_f32_16x16x4_f32(
            false, av, false, bv, (short)0, acc, false, false);
    }

    if (fullTile) {
        float* cp = C + (size_t)(mbase + half * 8) * FEAT + ncol;
#pragma unroll
        for (int r = 0; r < 8; ++r) cp[(size_t)r * FEAT] = acc[r];
    } else {
#pragma unroll
        for (int r = 0; r < 8; ++r) {
            int grow = mbase + half * 8 + r;
            if (grow < nrows) C[(size_t)grow * FEAT + ncol] = acc[r];
        }
    }
}

// ---------------------------------------------------------------------------
// Edge scatter: agg[col] += h[row] * dis[row]*dis[col].
// One wave per edge; each lane moves 4 contiguous floats (float4 gather,
// 4x global_atomic_add_f32 scatter). All buffers are L2 resident.
// ---------------------------------------------------------------------------
__global__ __launch_bounds__(256)
void edge_scatter_kernel(const float* __restrict__ h, const int* __restrict__ row,
                         const int* __restrict__ col, const float* __restrict__ dis,
                         float* __restrict__ agg, int n_edges) {
    int gid  = blockIdx.x * blockDim.x + threadIdx.x;
    int e    = gid >> 5;
    if (e >= n_edges) return;
    int lane = threadIdx.x & 31;

    int r = row[e];
    int c = col[e];
    float coef = dis[r] * dis[c];

    const float4 hv = *reinterpret_cast<const float4*>(h + (size_t)r * FEAT + lane * 4);
    float* dst = agg + (size_t)c * FEAT + lane * 4;
    atomicAdd(dst + 0, hv.x * coef);
    atomicAdd(dst + 1, hv.y * coef);
    atomicAdd(dst + 2, hv.z * coef);
    atomicAdd(dst + 3, hv.w * coef);
}

// ---------------------------------------------------------------------------
// Combine: out = agg + h*dis^2 + bias. Optionally accumulate per-feature
// sum / sum-of-squares (BN training stats) with one atomic pair per block.
// Block = 128 threads (thread == feature), each block walks a node chunk.
// Safe when out aliases h (each thread reads h[idx] before writing out[idx]).
// ---------------------------------------------------------------------------
template <bool STATS>
__global__ __launch_bounds__(128)
void combine_kernel(const float* __restrict__ agg, const float* __restrict__ h,
                    const float* __restrict__ dis, const float* __restrict__ bias,
                    float* __restrict__ out, float* __restrict__ sums,
                    int n_nodes, int chunk) {
    const int f  = threadIdx.x;
    const int n0 = blockIdx.x * chunk;
    const int n1 = min(n0 + chunk, n_nodes);
    const float b = bias[f];

    float s = 0.0f, s2 = 0.0f;
    for (int n = n0; n < n1; ++n) {
        float d = dis[n];
        size_t idx = (size_t)n * FEAT + f;
        float v = agg[idx] + h[idx] * (d * d) + b;
        out[idx] = v;
        if (STATS) { s += v; s2 += v * v; }
    }
    if (STATS) {
        atomicAdd(&sums[f], s);
        atomicAdd(&sums[FEAT + f], s2);
    }
}

// ---------------------------------------------------------------------------
// BN finalize: fold gamma/beta/mean/invstd into y = a*x + c (128 features).
// ---------------------------------------------------------------------------
__global__ __launch_bounds__(128)
void bn_finalize_kernel(const float* __restrict__ sums, const float* __restrict__ gamma,
                        const float* __restrict__ beta, float* __restrict__ bnA,
                        float* __restrict__ bnC, float inv_n) {
    int f = threadIdx.x;
    float mean = sums[f] * inv_n;
    float var  = sums[FEAT + f] * inv_n - mean * mean;
    float a    = rsqrtf(var + BN_EPS) * gamma[f];
    bnA[f] = a;
    bnC[f] = beta[f] - mean * a;
}

__global__ __launch_bounds__(256)
void bn_apply_kernel(const float* __restrict__ in, float* __restrict__ out,
                     const float* __restrict__ bnA, const float* __restrict__ bnC,
                     size_t total) {
    size_t i = ((size_t)blockIdx.x * blockDim.x + threadIdx.x) * 4;
    if (i >= total) return;
    int f = (int)(i & (FEAT - 1));
    float4 v = *reinterpret_cast<const float4*>(in + i);
    float4 a = *reinterpret_cast<const float4*>(bnA + f);
    float4 c = *reinterpret_cast<const float4*>(bnC + f);
    float4 o;
    o.x = v.x * a.x + c.x;
    o.y = v.y * a.y + c.y;
    o.z = v.z * a.z + c.z;
    o.w = v.w * a.w + c.w;
    *reinterpret_cast<float4*>(out + i) = o;
}

// ---------------------------------------------------------------------------
// Host-side launcher
// ---------------------------------------------------------------------------
extern "C" void kernel_launch(void* const* d_in, const int* in_sizes, int n_in,
                              void* d_out, int out_size, void* d_ws, size_t ws_size,
                              hipStream_t stream) {
    (void)n_in; (void)out_size; (void)ws_size;

    const float* X     = (const float*)d_in[0];   // [N, 64]
    const int*   ei    = (const int*)d_in[1];     // [2, E]
    const float* W1    = (const float*)d_in[2];   // [64, 128]
    const float* b1    = (const float*)d_in[3];   // [128]
    const float* gamma = (const float*)d_in[4];   // [128]
    const float* beta  = (const float*)d_in[5];   // [128]
    const float* W2    = (const float*)d_in[6];   // [128, 128]
    const float* b2    = (const float*)d_in[7];   // [128]
    float*       out   = (float*)d_out;           // [N, 128]

    const int N = in_sizes[0] / 64;               // 100000
    const int E = in_sizes[1] / 2;                // 1600000
    const int* row = ei;
    const int* col = ei + E;

    // workspace carve-up (256B aligned); d_out doubles as the third big buffer
    auto align256 = [](size_t x) { return (x + 255) & ~(size_t)255; };
    char* ws = (char*)d_ws;
    size_t nodeBuf = align256((size_t)N * FEAT * sizeof(float));   // 51.2 MB
    float* buf0 = (float*)(ws);                    // h1, later bn_out
    float* buf1 = (float*)(ws + nodeBuf);          // agg1, later agg2
    float* buf2 = out;                             // out1 (pre-BN), later h2, final out
    char*  tail = ws + 2 * nodeBuf;
    float* deg  = (float*)tail;                    tail += align256((size_t)N * sizeof(float));
    float* dis  = (float*)tail;                    tail += align256((size_t)N * sizeof(float));
    float* sums = (float*)tail;                    tail += align256(2 * FEAT * sizeof(float));
    float* bnA  = (float*)tail;                    tail += align256(FEAT * sizeof(float));
    float* bnC  = (float*)tail;

    const int mTiles = (N + 15) / 16;
    const int chunk  = 256;
    const int nChunkBlocks = (N + chunk - 1) / chunk;
    const size_t totalElems = (size_t)N * FEAT;

    // 1. zero deg / stats / agg
    hipMemsetAsync(deg,  0, (size_t)N * sizeof(float), stream);
    hipMemsetAsync(sums, 0, 2 * FEAT * sizeof(float), stream);
    hipMemsetAsync(buf1, 0, (size_t)N * FEAT * sizeof(float), stream);

    // 2. degrees + D^{-1/2}
    deg_kernel<<<(E + 255) / 256, 256, 0, stream>>>(col, deg, E);
    dis_kernel<<<(N + 255) / 256, 256, 0, stream>>>(deg, dis, N);

    // 3. layer 1: h1 = X @ W1   (WMMA fp32, K=64)
    gemm_wmma_kernel<64><<<mTiles, 256, 0, stream>>>(X, W1, buf0, N);

    // 4. scatter: agg1 += coef * h1[row]
    edge_scatter_kernel<<<(size_t)E * 32 / 256, 256, 0, stream>>>(buf0, row, col, dis, buf1, E);

    // 5. combine + BN statistics -> buf2 (pre-BN activations)
    combine_kernel<true><<<nChunkBlocks, 128, 0, stream>>>(buf1, buf0, dis, b1, buf2, sums, N, chunk);

    // 6. BN train-mode normalize: buf0 = a*buf2 + c
    bn_finalize_kernel<<<1, 128, 0, stream>>>(sums, gamma, beta, bnA, bnC, 1.0f / (float)N);
    bn_apply_kernel<<<(unsigned)((totalElems / 4 + 255) / 256), 256, 0, stream>>>(buf2, buf0, bnA, bnC, totalElems);

    // 7. layer 2: h2 = bn_out @ W2   (WMMA fp32, K=128); re-zero agg
    hipMemsetAsync(buf1, 0, (size_t)N * FEAT * sizeof(float), stream);
    gemm_wmma_kernel<128><<<mTiles, 256, 0, stream>>>(buf0, W2, buf2, N);

    // 8. scatter + final combine -> d_out (out aliases h2; safe per-thread RAW order)
    edge_scatter_kernel<<<(size_t)E * 32 / 256, 256, 0, stream>>>(buf2, row, col, dis, buf1, E);
    combine_kernel<false><<<nChunkBlocks, 128, 0, stream>>>(buf1, buf2, dis, b2, out, nullptr, N, chunk);
}